// Net_50732153700838
// MI455X (gfx1250) — compile-verified
//
#include <hip/hip_runtime.h>

typedef float v2f __attribute__((ext_vector_type(2)));
typedef float v8f __attribute__((ext_vector_type(8)));

#define NN   116
#define NP   128
#define HID  512
#define EDIM 5
#define KENC 570
#define KEP  576   // KENC padded to a multiple of 16

// ---------------------------------------------------------------------------
// Branch-free WMMA f32 GEMM: C = A @ B (+bias) (+relu), pad rows of C zeroed.
// Compile-time strides: A:[NP,LDA], B:[KK,512], C:[NP,512] -> all global loads
// use immediate offsets from a single base per operand.
// Grid: (NP/16, (512/16)/4), Block: (32,4). One wave per 16x16 output tile.
// Four independent accumulator chains hide V_WMMA_F32_16X16X4_F32 RAW latency.
// ---------------------------------------------------------------------------
template <int KK, int LDA>
__global__ void gemm16_wmma(const float* __restrict__ A,
                            const float* __restrict__ B,
                            float* __restrict__ C, int Mvalid,
                            const float* __restrict__ bias, int do_relu) {
    constexpr int LDB = HID, LDC = HID;
    const int lane = threadIdx.x;                            // 0..31
    const int tm   = blockIdx.x;                             // M tile
    const int tn   = blockIdx.y * blockDim.y + threadIdx.y;  // N tile
    const int hl   = lane >> 4;                              // lane half
    const int l16  = lane & 15;

    const int arow = tm * 16 + l16;
    const int bcol = tn * 16 + l16;

    // A 16x4 f32 fragment: lanes 0-15 hold K=k0,k0+1; lanes 16-31 hold k0+2,k0+3
    const float* ap = A + arow * LDA + hl * 2;
    // B 4x16 f32 fragment: VGPR v, half h -> row k0 + 2h + v, col = lane&15
    const float* bp = B + (hl * 2) * LDB + bcol;

    v8f acc0 = {}, acc1 = {}, acc2 = {}, acc3 = {};
    #pragma unroll 2
    for (int k0 = 0; k0 < KK; k0 += 16) {
        v2f a0 = *(const v2f*)(ap + 0);      // global_load_b64, imm offsets
        v2f a1 = *(const v2f*)(ap + 4);
        v2f a2 = *(const v2f*)(ap + 8);
        v2f a3 = *(const v2f*)(ap + 12);
        v2f b0 = { bp[0 * LDB],  bp[1 * LDB]  };
        v2f b1 = { bp[4 * LDB],  bp[5 * LDB]  };
        v2f b2 = { bp[8 * LDB],  bp[9 * LDB]  };
        v2f b3 = { bp[12 * LDB], bp[13 * LDB] };
        __builtin_prefetch(bp + 32 * LDB, 0, 1);   // global_prefetch_b8
        acc0 = __builtin_amdgcn_wmma_f32_16x16x4_f32(
            false, a0, false, b0, (short)0, acc0, false, false);
        acc1 = __builtin_amdgcn_wmma_f32_16x16x4_f32(
            false, a1, false, b1, (short)0, acc1, false, false);
        acc2 = __builtin_amdgcn_wmma_f32_16x16x4_f32(
            false, a2, false, b2, (short)0, acc2, false, false);
        acc3 = __builtin_amdgcn_wmma_f32_16x16x4_f32(
            false, a3, false, b3, (short)0, acc3, false, false);
        ap += 16;
        bp += 16 * LDB;
    }

    const float bv = bias ? bias[bcol] : 0.0f;
    #pragma unroll
    for (int v = 0; v < 8; ++v) {
        // C/D layout: VGPR v, lanes 0-15 -> M=v, lanes 16-31 -> M=v+8
        const int row = tm * 16 + v + hl * 8;
        float val = (acc0[v] + acc1[v]) + (acc2[v] + acc3[v]) + bv;
        if (do_relu) val = fmaxf(val, 0.0f);
        if (row >= Mvalid) val = 0.0f;       // keep pads clean for next GEMM
        C[row * LDC + bcol] = val;
    }
}

// ---------------------------------------------------------------------------
// Zero-padded staging copy: dst[r,c] = (r<srows && c<scols) ? src[r,c] : 0
// ---------------------------------------------------------------------------
__global__ void pad_copy_kernel(const float* __restrict__ src, int srows, int scols,
                                float* __restrict__ dst, int drows, int dcols) {
    int idx = blockIdx.x * blockDim.x + threadIdx.x;
    if (idx >= drows * dcols) return;
    int r = idx / dcols, c = idx - r * dcols;
    dst[idx] = (r < srows && c < scols) ? src[r * scols + c] : 0.0f;
}

__device__ __forceinline__ int edge_id(int a, int b) {   // requires a < b
    return a * (2 * NN - a - 1) / 2 + (b - a - 1);       // triu_indices(N,1) order
}

// Dense padded adjacency (gather form), fused with w1 = edge_attr . nc1_p:
// Ad[i,j] = dot(edge_attr[e(i,j)], p) for i!=j (<116), else 0
__global__ void buildA1_kernel(const float* __restrict__ edge_attr,
                               const float* __restrict__ p,
                               float* __restrict__ Ad) {
    int idx = blockIdx.x * blockDim.x + threadIdx.x;
    if (idx >= NP * NP) return;
    int i = idx >> 7, j = idx & (NP - 1);
    float v = 0.0f;
    if (i < NN && j < NN && i != j) {
        int a = i < j ? i : j, b = i < j ? j : i;
        int e = edge_id(a, b);
        #pragma unroll
        for (int d = 0; d < EDIM; ++d) v += edge_attr[e * EDIM + d] * p[d];
    }
    Ad[idx] = v;
}

// Ad[i,j] = w3[e(i,j)] for i!=j (<116), else 0
__global__ void buildA2_kernel(const float* __restrict__ w3,
                               float* __restrict__ Ad) {
    int idx = blockIdx.x * blockDim.x + threadIdx.x;
    if (idx >= NP * NP) return;
    int i = idx >> 7, j = idx & (NP - 1);
    float v = 0.0f;
    if (i < NN && j < NN && i != j) {
        int a = i < j ? i : j, b = i < j ? j : i;
        v = w3[edge_id(a, b)];
    }
    Ad[idx] = v;
}

// w2[n] = x[n,:] . p[:]   (HID dot)
__global__ void node_w_kernel(const float* __restrict__ X,
                              const float* __restrict__ p,
                              float* __restrict__ w) {
    int n = blockIdx.x * blockDim.x + threadIdx.x;
    if (n >= NN) return;
    float s = 0.0f;
    for (int k = 0; k < HID; ++k) s += X[n * HID + k] * p[k];
    w[n] = s;
}

// Mn[e,:] = (relu(edge_attr[e,:]) @ ecW) / max(0, w2[i], w2[j])
__global__ void edge_mn_kernel(const float* __restrict__ edge_attr,
                               const float* __restrict__ ecW,   // [EDIM,EDIM]
                               const int* __restrict__ ei,
                               const float* __restrict__ w2,
                               float* __restrict__ Mn, int E) {
    int e = blockIdx.x * blockDim.x + threadIdx.x;
    if (e >= E) return;
    float he[EDIM];
    #pragma unroll
    for (int d = 0; d < EDIM; ++d) he[d] = fmaxf(edge_attr[e * EDIM + d], 0.0f);
    int i = ei[e], j = ei[E + e];
    float cm = fmaxf(0.0f, fmaxf(w2[i], w2[j]));  // col max incl. structural zeros
    #pragma unroll
    for (int d = 0; d < EDIM; ++d) {
        float m = 0.0f;
        #pragma unroll
        for (int c = 0; c < EDIM; ++c) m += he[c] * ecW[c * EDIM + d];
        Mn[e * EDIM + d] = m / cm;                 // IEEE semantics match reference
    }
}

// S[n,:] = sum over edges incident to n of Mn (deterministic, no atomics)
__global__ void node_S_kernel(const float* __restrict__ Mn, float* __restrict__ S) {
    int n = blockIdx.x * blockDim.x + threadIdx.x;
    if (n >= NN) return;
    float s[EDIM] = {0.f, 0.f, 0.f, 0.f, 0.f};
    for (int m = 0; m < NN; ++m) {
        if (m == n) continue;
        int e = (m < n) ? edge_id(m, n) : edge_id(n, m);
        #pragma unroll
        for (int d = 0; d < EDIM; ++d) s[d] += Mn[e * EDIM + d];
    }
    #pragma unroll
    for (int d = 0; d < EDIM; ++d) S[n * EDIM + d] = s[d];
}

// Edge-conv row, folded into w3[e] = relu(row) . nc2_p
__global__ void edge_out_kernel(const int* __restrict__ ei,
                                const float* __restrict__ w2,
                                const float* __restrict__ S,
                                const float* __restrict__ Mn,
                                const float* __restrict__ ecb,   // ec1_b
                                const float* __restrict__ nc2p,  // nc2_p
                                float* __restrict__ w3, int E) {
    int e = blockIdx.x * blockDim.x + threadIdx.x;
    if (e >= E) return;
    int i = ei[e], j = ei[E + e];
    float wi = w2[i], wj = w2[j];
    float acc = 0.0f;
    #pragma unroll
    for (int d = 0; d < EDIM; ++d) {
        float mn = Mn[e * EDIM + d];
        float v = wi * (S[i * EDIM + d] - mn) + wj * (S[j * EDIM + d] - mn) + ecb[d];
        v = fmaxf(v, 0.0f);
        acc += v * nc2p[d];
    }
    w3[e] = acc;
}

// mean-pool over 116 rows -> [512], linear -> [4], log_softmax -> d_out
__global__ void finalize_kernel(const float* __restrict__ X,      // [NP,HID]
                                const float* __restrict__ linW,   // [HID,4]
                                const float* __restrict__ linb,   // [4]
                                float* __restrict__ out) {
    __shared__ float pooled[HID];
    __shared__ float logits[4];
    int c = threadIdx.x;   // 0..511
    float s = 0.0f;
    for (int r = 0; r < NN; ++r) s += X[r * HID + c];
    pooled[c] = s * (1.0f / (float)NN);
    __syncthreads();
    if (c < 4) {
        float l = linb[c];
        for (int k = 0; k < HID; ++k) l += pooled[k] * linW[k * 4 + c];
        logits[c] = l;
    }
    __syncthreads();
    if (c == 0) {
        float m = logits[0];
        #pragma unroll
        for (int o = 1; o < 4; ++o) m = fmaxf(m, logits[o]);
        float sum = 0.0f;
        #pragma unroll
        for (int o = 0; o < 4; ++o) sum += __expf(logits[o] - m);
        float lse = __logf(sum);
        #pragma unroll
        for (int o = 0; o < 4; ++o) out[o] = logits[o] - m - lse;
    }
}

// ---------------------------------------------------------------------------
extern "C" void kernel_launch(void* const* d_in, const int* in_sizes, int n_in,
                              void* d_out, int out_size, void* d_ws, size_t ws_size,
                              hipStream_t stream) {
    const float* edge_attr = (const float*)d_in[0];
    const float* enc_raw   = (const float*)d_in[1];   // [116, 570]
    const float* W_enc     = (const float*)d_in[2];   // [570, 512]
    const float* b_enc     = (const float*)d_in[3];
    const float* nc1_W     = (const float*)d_in[4];
    const float* nc1_p     = (const float*)d_in[5];
    const float* nc1_b     = (const float*)d_in[6];
    const float* ec1_W     = (const float*)d_in[7];
    const float* ec1_p     = (const float*)d_in[8];
    const float* ec1_b     = (const float*)d_in[9];
    const float* nc2_W     = (const float*)d_in[10];
    const float* nc2_p     = (const float*)d_in[11];
    const float* nc2_b     = (const float*)d_in[12];
    const float* lin_W     = (const float*)d_in[13];
    const float* lin_b     = (const float*)d_in[14];
    const int*   edge_idx  = (const int*)d_in[15];    // [2, E]
    const int    E         = in_sizes[0] / EDIM;      // 6670

    float* ws    = (float*)d_ws;
    float* encP  = ws;                           // NP * KEP    (padded enc_raw)
    float* WencP = encP  + NP * KEP;             // KEP * HID   (padded W_enc)
    float* x0    = WencP + (size_t)KEP * HID;    // NP * HID
    float* x1    = x0    + NP * HID;             // NP * HID
    float* tmp   = x1    + NP * HID;             // NP * HID
    float* out2  = tmp   + NP * HID;             // NP * HID
    float* Ad    = out2  + NP * HID;             // NP * NP
    float* w2    = Ad    + NP * NP;              // NP
    float* w3    = w2    + NP;                   // E
    float* Mn    = w3    + E;                    // E * EDIM
    float* S     = Mn    + (size_t)E * EDIM;     // NP * EDIM

    const dim3 gblk(32, 4);                      // 4 waves, one 16x16 tile each
    const dim3 ggrd(NP / 16, (HID / 16) / 4);
    const int  eb = 256, eg = (E + eb - 1) / eb;

    // 0) stage zero-padded operands for the encoding GEMM
    pad_copy_kernel<<<(NP * KEP + 255) / 256, 256, 0, stream>>>(
        enc_raw, NN, KENC, encP, NP, KEP);
    pad_copy_kernel<<<(KEP * HID + 255) / 256, 256, 0, stream>>>(
        W_enc, KENC, HID, WencP, KEP, HID);
    // 1) x0 = enc_raw @ W_enc + b_enc
    gemm16_wmma<KEP, KEP><<<ggrd, gblk, 0, stream>>>(encP, WencP,
                                                     x0, NN, b_enc, 0);
    // 2) dense A1 (fused w1 = edge_attr . nc1_p)
    buildA1_kernel<<<(NP * NP + 255) / 256, 256, 0, stream>>>(edge_attr, nc1_p, Ad);
    // 3) tmp = x0 @ nc1_W
    gemm16_wmma<HID, HID><<<ggrd, gblk, 0, stream>>>(x0, nc1_W,
                                                     tmp, NN, nullptr, 0);
    // 4) x1 = relu(A1 @ tmp + nc1_b)
    gemm16_wmma<NP, NP><<<ggrd, gblk, 0, stream>>>(Ad, tmp,
                                                   x1, NN, nc1_b, 1);
    // 5) w2[n] = x1[n] . ec1_p
    node_w_kernel<<<1, 128, 0, stream>>>(x1, ec1_p, w2);
    // 6) Mn[e] = (relu(edge_attr) @ ec1_W) / colmax
    edge_mn_kernel<<<eg, eb, 0, stream>>>(edge_attr, ec1_W, edge_idx, w2, Mn, E);
    // 7) S[n] = sum of Mn over incident edges
    node_S_kernel<<<1, 128, 0, stream>>>(Mn, S);
    // 8) w3[e] = relu(edge_conv row) . nc2_p
    edge_out_kernel<<<eg, eb, 0, stream>>>(edge_idx, w2, S, Mn, ec1_b, nc2_p, w3, E);
    // 9) dense A2 from w3
    buildA2_kernel<<<(NP * NP + 255) / 256, 256, 0, stream>>>(w3, Ad);
    // 10) tmp = x1 @ nc2_W    (relu(x1) == x1)
    gemm16_wmma<HID, HID><<<ggrd, gblk, 0, stream>>>(x1, nc2_W,
                                                     tmp, NN, nullptr, 0);
    // 11) out2 = A2 @ tmp + nc2_b
    gemm16_wmma<NP, NP><<<ggrd, gblk, 0, stream>>>(Ad, tmp,
                                                   out2, NN, nc2_b, 0);
    // 12) mean-pool -> linear -> log_softmax
    finalize_kernel<<<1, HID, 0, stream>>>(out2, lin_W, lin_b, (float*)d_out);
}